// RoIPoolingLayer_32246614459255
// MI455X (gfx1250) — compile-verified
//
#include <hip/hip_runtime.h>
#include <hip/hip_bf16.h>
#include <math.h>

// RoI max-pooling, faithful to the TF/JAX reference:
//   step = h / 7 (fp32); cell = [int(p*step), int((p+1)*step)) truncated,
//   last cell ends at h; cell sizes clamped >= 1; rows/cols clipped to map.
//
// Layouts:
//   x_maps: (B, H, W, C) fp32, C innermost -> contiguous 2KB per (row,col)
//   x_rois: (B, R, 4) int32 = (y, x, h, w)
//   out:    (B, R, 7, 7, C) fp32
//
// Grid (49, R, B): roi/batch indices come straight from blockIdx.y/z (TTMP
// hardware registers) — no runtime integer division in the prologue; only a
// constant /7 remains (one mul-hi). One 128-thread block (4 wave32 waves)
// per output cell; each thread owns 4 channels (global_load_b128, fully
// coalesced). Window <= 4x4 positions -> <= 16 independent b128 loads per
// thread feeding v_max chains. Output streamed with non-temporal stores
// (TH=NT) so the 12.8MB of write-once data does not evict the 10MB map
// working set from the 192MB L2.

typedef float v4f __attribute__((ext_vector_type(4)));
typedef int   v4i __attribute__((ext_vector_type(4)));

namespace {
constexpr int H = 50;
constexpr int W = 50;
constexpr int C = 512;
constexpr int P = 7;          // POOL_SIZE
constexpr int TPB = C / 4;    // 128 threads, one vec4 each
}

__global__ __launch_bounds__(TPB, 8)
void roi_pool_kernel(const float* __restrict__ maps,
                     const int*   __restrict__ rois,
                     float*       __restrict__ out,
                     int R)
{
    // (px, py) from a constant-divisor decode; (r, b) free from blockIdx.y/z.
    const int cell = blockIdx.x;        // [0, 49)
    const int py = cell / P;            // constant /7 -> mul-hi
    const int px = cell - py * P;
    const int r  = blockIdx.y;
    const int b  = blockIdx.z;

    // ROI descriptor (uniform across block -> scalar loads)
    const v4i roi = *(const v4i*)(rois + (size_t)(b * R + r) * 4);
    const int ry = roi.x, rx = roi.y, rh = roi.z, rw = roi.w;

    // fp32 boundary math, truncation toward zero — bit-exact vs reference
    const float y_step = (float)rh / (float)P;
    const float x_step = (float)rw / (float)P;

    const int y_start = (int)((float)py * y_step);
    const int y_end   = (py + 1 < P) ? (int)((float)(py + 1) * y_step) : rh;
    const int y_size  = max(y_end - y_start, 1);

    const int x_start = (int)((float)px * x_step);
    const int x_end   = (px + 1 < P) ? (int)((float)(px + 1) * x_step) : rw;
    const int x_size  = max(x_end - x_start, 1);

    const int c = threadIdx.x * 4;
    const float* __restrict__ base = maps + (size_t)b * H * W * C;

    // Kick off each window row's fetch early (global_prefetch_b8).
    #pragma unroll 4
    for (int ky = 0; ky < y_size; ++ky) {
        int row = min(max(ry + y_start + ky, 0), H - 1);
        int col = min(max(rx + x_start, 0), W - 1);
        __builtin_prefetch(base + ((size_t)row * W + col) * C + c, 0, 3);
    }

    v4f acc = { -INFINITY, -INFINITY, -INFINITY, -INFINITY };

    #pragma unroll 4
    for (int ky = 0; ky < y_size; ++ky) {
        const int row = min(max(ry + y_start + ky, 0), H - 1);
        const float* __restrict__ rowp = base + (size_t)row * W * C + c;
        #pragma unroll 4
        for (int kx = 0; kx < x_size; ++kx) {
            const int col = min(max(rx + x_start + kx, 0), W - 1);
            const v4f v = *(const v4f*)(rowp + (size_t)col * C);
            acc.x = fmaxf(acc.x, v.x);
            acc.y = fmaxf(acc.y, v.y);
            acc.z = fmaxf(acc.z, v.z);
            acc.w = fmaxf(acc.w, v.w);
        }
    }

    // (b, r, py, px, c) — streamed once, never re-read: non-temporal store.
    float* op = out + ((((size_t)(b * R + r)) * P * P) + (size_t)(py * P + px)) * C + c;
    __builtin_nontemporal_store(acc, (v4f*)op);
}

extern "C" void kernel_launch(void* const* d_in, const int* in_sizes, int n_in,
                              void* d_out, int out_size, void* d_ws, size_t ws_size,
                              hipStream_t stream)
{
    const float* maps = (const float*)d_in[0];
    const int*   rois = (const int*)d_in[1];
    float*       out  = (float*)d_out;

    const int B = in_sizes[0] / (H * W * C);           // 2
    const int R = in_sizes[1] / (4 * (B > 0 ? B : 1)); // 64

    roi_pool_kernel<<<dim3(P * P, R, B), dim3(TPB), 0, stream>>>(maps, rois, out, R);
}